// forward_warp_1726576856262
// MI455X (gfx1250) — compile-verified
//
#include <hip/hip_runtime.h>

// Problem constants (match reference: B,C,H,W = 16,3,512,960)
#define BB 16
#define CC 3
#define HH 512
#define WW 960
#define HWP (HH * WW)          // 491,520 pixels per plane
#define NPIX (BB * HWP)        // 7,864,320 source pixels
#define NOUT (BB * CC * HWP)   // 23,592,960 output floats

typedef __attribute__((ext_vector_type(2))) float v2f;
typedef __attribute__((ext_vector_type(4))) float v4f;

// Non-returning, agent-scope, relaxed f32 atomic add -> global_atomic_add_f32
__device__ __forceinline__ void atom_add_f32(float* p, float v) {
    __hip_atomic_fetch_add(p, v, __ATOMIC_RELAXED, __HIP_MEMORY_SCOPE_AGENT);
}

// ---------------------------------------------------------------------------
// Bilinear forward splat. One thread per source pixel.
// Grid: (WW/320, HH, BB), block: 320 (10 wave32s) -> no integer divisions:
//   w = blockIdx.x*320 + tid, h = blockIdx.y, b = blockIdx.z.
//  - flow read as coalesced 64b non-temporal load (read-once)
//  - 3 channel values read as coalesced 32b non-temporal loads
//  - 12 fire-and-forget global_atomic_add_f32 into the L2-resident output
// (Defined first in the module so the disasm snippet shows this kernel.)
// ---------------------------------------------------------------------------
__global__ __launch_bounds__(320) void fw_splat_kernel(
        const float* __restrict__ im0,
        const v2f*  __restrict__ flow,
        float*      __restrict__ out) {
    const int w = blockIdx.x * 320 + threadIdx.x;   // 0 .. 959
    const int h = blockIdx.y;                       // 0 .. 511
    const int b = blockIdx.z;                       // 0 .. 15

    const int s = h * WW + w;                       // pixel within image
    const int p = b * HWP + s;                      // global pixel index

    // flow[b,h,w,{x,y}] : contiguous float2 per pixel -> global_load_b64 (NT)
    const v2f f = __builtin_nontemporal_load(&flow[p]);

    const float x = (float)w + f.x;
    const float y = (float)h + f.y;
    const float xf = floorf(x);
    const float yf = floorf(y);

    // Same validity condition as reference, evaluated in float before any
    // int cast (x_f >= 0 && x_f+1 < W && y_f >= 0 && y_f+1 < H).
    const bool valid = (xf >= 0.0f) & (xf + 1.0f < (float)WW) &
                       (yf >= 0.0f) & (yf + 1.0f < (float)HH);

    // Channel loads: coalesced within each plane, read-once -> non-temporal.
    const float* src = im0 + (size_t)b * (CC * HWP) + s;
    const float v0 = __builtin_nontemporal_load(src);
    const float v1 = __builtin_nontemporal_load(src + HWP);
    const float v2 = __builtin_nontemporal_load(src + 2 * HWP);

    if (!valid) return;

    const int xi = (int)xf;          // safe: valid guarantees in-range
    const int yi = (int)yf;

    const float ax = x - xf;         // x - x_f
    const float ay = y - yf;         // y - y_f
    const float bx = 1.0f - ax;      // x_c - x
    const float by = 1.0f - ay;      // y_c - y

    const float k_nw = bx * by;
    const float k_ne = ax * by;
    const float k_sw = bx * ay;
    const float k_se = ax * ay;

    float* dst = out + (size_t)b * (CC * HWP) + (size_t)yi * WW + xi;

    // 12 non-returning agent-scope atomics; L2-resident RMW.
    atom_add_f32(dst,              k_nw * v0);
    atom_add_f32(dst + 1,          k_ne * v0);
    atom_add_f32(dst + WW,         k_sw * v0);
    atom_add_f32(dst + WW + 1,     k_se * v0);
    dst += HWP;
    atom_add_f32(dst,              k_nw * v1);
    atom_add_f32(dst + 1,          k_ne * v1);
    atom_add_f32(dst + WW,         k_sw * v1);
    atom_add_f32(dst + WW + 1,     k_se * v1);
    dst += HWP;
    atom_add_f32(dst,              k_nw * v2);
    atom_add_f32(dst + 1,          k_ne * v2);
    atom_add_f32(dst + WW,         k_sw * v2);
    atom_add_f32(dst + WW + 1,     k_se * v2);
}

// ---------------------------------------------------------------------------
// Zero the output. Regular-temporal stores so the 94.4 MB output becomes
// resident in the 192 MB L2, where all subsequent atomics will hit.
// NOUT/4 = 5,898,240 float4s; 23040 blocks * 256 threads covers it exactly.
// ---------------------------------------------------------------------------
__global__ __launch_bounds__(256) void fw_zero_kernel(v4f* __restrict__ out) {
    int i = blockIdx.x * blockDim.x + threadIdx.x;
    v4f z = {0.f, 0.f, 0.f, 0.f};
    out[i] = z;
}

extern "C" void kernel_launch(void* const* d_in, const int* in_sizes, int n_in,
                              void* d_out, int out_size, void* d_ws, size_t ws_size,
                              hipStream_t stream) {
    const float* im0  = (const float*)d_in[0];   // [B,C,H,W] f32
    const v2f*   flow = (const v2f*)d_in[1];     // [B,H,W,2] f32
    float*       out  = (float*)d_out;           // [B,C,H,W] f32

    // 1) zero output (poisoned by harness; must re-zero deterministically)
    fw_zero_kernel<<<NOUT / 4 / 256, 256, 0, stream>>>((v4f*)out);

    // 2) splat (same stream -> ordered after the zero pass)
    dim3 grid(WW / 320, HH, BB);
    fw_splat_kernel<<<grid, 320, 0, stream>>>(im0, flow, out);
}